// SGC_25847113187632
// MI455X (gfx1250) — compile-verified
//
#include <hip/hip_runtime.h>

#define N_NODES    50000
#define N_EDGES    1600000
#define N_FEAT     128
#define N_CLASSES  64

typedef float v2f __attribute__((ext_vector_type(2)));
typedef float v8f __attribute__((ext_vector_type(8)));

// ---------------------------------------------------------------------------
// 1) degree init: deg[i] = 1.0 (self loop contribution)
// ---------------------------------------------------------------------------
__global__ void sgc_deg_init(float* __restrict__ deg, int n) {
    int i = blockIdx.x * blockDim.x + threadIdx.x;
    if (i < n) deg[i] = 1.0f;
}

// ---------------------------------------------------------------------------
// 2) degree accumulate: deg[rows[e]] += A_data[e]
// ---------------------------------------------------------------------------
__global__ void sgc_deg_accum(const int* __restrict__ idx,
                              const float* __restrict__ adata,
                              float* __restrict__ deg, int nE) {
    int e = blockIdx.x * blockDim.x + threadIdx.x;
    if (e < nE) {
        int r = idx[2 * e];
        unsafeAtomicAdd(&deg[r], adata[e]);
    }
}

// ---------------------------------------------------------------------------
// 3) invd[i] = 1/sqrt(deg[i])  (in place)
// ---------------------------------------------------------------------------
__global__ void sgc_deg_invsqrt(float* __restrict__ deg, int n) {
    int i = blockIdx.x * blockDim.x + threadIdx.x;
    if (i < n) deg[i] = 1.0f / sqrtf(deg[i]);
}

// ---------------------------------------------------------------------------
// 4) per-layer init: xout[i,:] = invd[i]^2 * xin[i,:]   (self-loop edge)
//    one thread per float4 chunk (128 feats = 32 chunks per node)
// ---------------------------------------------------------------------------
__global__ void sgc_self_init(const float* __restrict__ invd,
                              const float* __restrict__ xin,
                              float* __restrict__ xout, int n) {
    int gid = blockIdx.x * blockDim.x + threadIdx.x;
    if (gid >= n * (N_FEAT / 4)) return;
    int i = gid >> 5;          // node
    int q = gid & 31;          // float4 chunk
    float s = invd[i] * invd[i];
    float4 m = ((const float4*)(xin + (size_t)i * N_FEAT))[q];
    float4 o;
    o.x = s * m.x; o.y = s * m.y; o.z = s * m.z; o.w = s * m.w;
    ((float4*)(xout + (size_t)i * N_FEAT))[q] = o;
}

// ---------------------------------------------------------------------------
// 5) edge SpMM: one wave32 per edge; lane handles a float4 feature chunk.
//    xout[r,:] += A_data[e]*invd[r]*invd[c] * xin[c,:]
//    X (25.6 MB) is L2-resident on MI455X (192 MB L2) -> gathers stay on chip.
// ---------------------------------------------------------------------------
__global__ void sgc_spmm_edges(const int* __restrict__ idx,
                               const float* __restrict__ adata,
                               const float* __restrict__ invd,
                               const float* __restrict__ xin,
                               float* __restrict__ xout, int nE) {
    int gid  = blockIdx.x * blockDim.x + threadIdx.x;
    int e    = gid >> 5;   // wave32 per edge
    int lane = gid & 31;
    if (e >= nE) return;
    int r = idx[2 * e];
    int c = idx[2 * e + 1];
    float v = adata[e] * invd[r] * invd[c];
    float4 m = ((const float4*)(xin + (size_t)c * N_FEAT))[lane];
    float* dst = xout + (size_t)r * N_FEAT + lane * 4;
    unsafeAtomicAdd(dst + 0, v * m.x);
    unsafeAtomicAdd(dst + 1, v * m.y);
    unsafeAtomicAdd(dst + 2, v * m.z);
    unsafeAtomicAdd(dst + 3, v * m.w);
}

// ---------------------------------------------------------------------------
// 6) final linear via fp32 WMMA: out[50000x64] = Xp[50000x128] * W^T + bias
//    One wave per 16x16 output tile; 32 x V_WMMA_F32_16X16X4_F32 over K=128.
//    Layouts per CDNA5 ISA 7.12.2 (32-bit A 16x4 / B 4x16 / C 16x16).
// ---------------------------------------------------------------------------
__global__ void sgc_wmma_linear(const float* __restrict__ Xp,
                                const float* __restrict__ W,
                                const float* __restrict__ bias,
                                float* __restrict__ out, int M) {
    const int lane = threadIdx.x & 31;
    const int wave = (blockIdx.x * blockDim.x + threadIdx.x) >> 5;
    const int tilesN = N_CLASSES / 16;              // 4
    const int tm = wave / tilesN;
    const int tn = wave % tilesN;
    if (tm * 16 >= M) return;                       // wave-uniform: EXEC all-1s below

    const int mrow  = tm * 16 + (lane & 15);        // A: M = lane%16
    const int ncol  = tn * 16 + (lane & 15);        // B: N = lane%16
    const int khalf = (lane >> 4) * 2;              // K base within 4-wide step

    const float* arow = Xp + (size_t)mrow * N_FEAT + khalf;  // A[m][khalf + v]
    const float* brow = W  + (size_t)ncol * N_FEAT + khalf;  // B[k][n] = W[n][k]

    v8f acc = {};
    for (int kb = 0; kb < N_FEAT; kb += 4) {
        v2f a, b;
        a[0] = arow[kb];     a[1] = arow[kb + 1];
        b[0] = brow[kb];     b[1] = brow[kb + 1];
        acc = __builtin_amdgcn_wmma_f32_16x16x4_f32(
            /*neg_a=*/false, a, /*neg_b=*/false, b,
            /*c_mod=*/(short)0, acc, /*reuse_a=*/false, /*reuse_b=*/false);
    }

    // C/D layout: VGPR v, lanes 0-15 -> M=v ; lanes 16-31 -> M=v+8 ; N=lane%16
    const int baseM = tm * 16 + 8 * (lane >> 4);
    const int col   = tn * 16 + (lane & 15);
    const float bv  = bias[col];
#pragma unroll
    for (int v = 0; v < 8; ++v) {
        out[(size_t)(baseM + v) * N_CLASSES + col] = acc[v] + bv;
    }
}

// ---------------------------------------------------------------------------
// host launcher
// ---------------------------------------------------------------------------
extern "C" void kernel_launch(void* const* d_in, const int* in_sizes, int n_in,
                              void* d_out, int out_size, void* d_ws, size_t ws_size,
                              hipStream_t stream) {
    const float* X      = (const float*)d_in[0];   // 50000*128
    const float* A_data = (const float*)d_in[1];   // 1600000
    const int*   A_idx  = (const int*)d_in[2];     // 1600000*2
    const float* Wgt    = (const float*)d_in[3];   // 64*128
    const float* bias   = (const float*)d_in[4];   // 64
    float* out = (float*)d_out;                    // 50000*64

    // workspace layout
    float* deg = (float*)d_ws;                     // 50000 f32 (becomes invd)
    float* xa  = deg + N_NODES;                    // 50000*128 f32
    float* xb  = xa + (size_t)N_NODES * N_FEAT;    // 50000*128 f32

    const int B = 256;

    // degree + normalization
    sgc_deg_init<<<(N_NODES + B - 1) / B, B, 0, stream>>>(deg, N_NODES);
    sgc_deg_accum<<<(N_EDGES + B - 1) / B, B, 0, stream>>>(A_idx, A_data, deg, N_EDGES);
    sgc_deg_invsqrt<<<(N_NODES + B - 1) / B, B, 0, stream>>>(deg, N_NODES);

    const int chunks      = N_NODES * (N_FEAT / 4);         // threads for self-init
    const int edgeThreads = N_EDGES * 32;                   // wave per edge

    // layer 1: X -> xa
    sgc_self_init<<<(chunks + B - 1) / B, B, 0, stream>>>(deg, X, xa, N_NODES);
    sgc_spmm_edges<<<(edgeThreads + B - 1) / B, B, 0, stream>>>(A_idx, A_data, deg, X, xa, N_EDGES);

    // layer 2: xa -> xb
    sgc_self_init<<<(chunks + B - 1) / B, B, 0, stream>>>(deg, xa, xb, N_NODES);
    sgc_spmm_edges<<<(edgeThreads + B - 1) / B, B, 0, stream>>>(A_idx, A_data, deg, xa, xb, N_EDGES);

    // final dense linear via fp32 WMMA
    const int tiles      = (N_NODES / 16) * (N_CLASSES / 16);   // 12500 waves
    const int gemmThread = tiles * 32;
    sgc_wmma_linear<<<(gemmThread + B - 1) / B, B, 0, stream>>>(xb, Wgt, bias, out, N_NODES);
}